// MambaModel_75204877353019
// MI455X (gfx1250) — compile-verified
//
#include <hip/hip_runtime.h>
#include <math.h>

// MambaModel on MI455X (gfx1250, wave32).
// Pipeline:
//   K1 k_gate_gemm : u  = gelu(x Wu^T + bu) * sigmoid(x Wg^T + bg)   [f32 WMMA dual-GEMM]
//   K2 k_gemm_nt   : uB = u  Bm^T                                    [f32 WMMA GEMM]
//   K3 k_scan_chunk: per-channel recurrence s = a_h s + uB (A is diagonal: S4D/Mamba),
//                    chunked over T for parallelism (local scans + chunk-end carries)
//   K4 k_scan_fix  : apply decay-weighted carry prefix across chunks
//   K5 k_gemm_nt   : out = S Cm^T + D                                [f32 WMMA GEMM + bias]

typedef __attribute__((ext_vector_type(2))) float v2f;
typedef __attribute__((ext_vector_type(8))) float v8f;

#define WMMA4(A_, B_, C_) \
  __builtin_amdgcn_wmma_f32_16x16x4_f32(false, (A_), false, (B_), (short)0, (C_), false, false)

static constexpr int kB = 8;
static constexpr int kT = 1024;
static constexpr int kD = 1024;        // K dim (D_IN)
static constexpr int kH = 1024;        // N dim (H)
static constexpr int kM = kB * kT;     // 8192 GEMM rows
static constexpr int kC = 8;           // scan chunks
static constexpr int kL = kT / kC;     // 128 steps per chunk

// ---------------------------------------------------------------------------
// Kernel 1: fused dual GEMM + gate.  Wave computes a 32x32 tile of u.
// ---------------------------------------------------------------------------
__device__ __forceinline__ float gate_act(float up, float gp) {
  // exact gelu (erf form) * sigmoid
  float g = 0.5f * up * (1.0f + erff(up * 0.7071067811865476f));
  float s = 1.0f / (1.0f + expf(-gp));
  return g * s;
}

__global__ __launch_bounds__(256) void k_gate_gemm(
    const float* __restrict__ X, const float* __restrict__ Wu,
    const float* __restrict__ bu, const float* __restrict__ Wg,
    const float* __restrict__ bg, float* __restrict__ U) {
  const int wave = (blockIdx.x * blockDim.x + threadIdx.x) >> 5;
  const int lane = threadIdx.x & 31;
  const int tilesM = kM / 32;            // 256
  const int tn = wave / tilesM;          // 0..31
  const int tm = wave % tilesM;          // 0..255
  const int m0 = tm * 32, n0 = tn * 32;
  const int l = lane & 15, hi = lane >> 4;

  const float* pA0 = X  + (size_t)(m0 + l) * kD + 2 * hi;      // A rows m0..m0+15
  const float* pA1 = pA0 + (size_t)16 * kD;                    // A rows m0+16..31
  const float* pU0 = Wu + (size_t)(n0 + l) * kD + 2 * hi;
  const float* pU1 = pU0 + (size_t)16 * kD;
  const float* pG0 = Wg + (size_t)(n0 + l) * kD + 2 * hi;
  const float* pG1 = pG0 + (size_t)16 * kD;

  v8f u00 = {}, u01 = {}, u10 = {}, u11 = {};
  v8f g00 = {}, g01 = {}, g10 = {}, g11 = {};

#pragma unroll 4
  for (int k0 = 0; k0 < kD; k0 += 4) {
    v2f a0  = *(const v2f*)(pA0 + k0);
    v2f a1  = *(const v2f*)(pA1 + k0);
    v2f wu0 = *(const v2f*)(pU0 + k0);
    v2f wu1 = *(const v2f*)(pU1 + k0);
    v2f wg0 = *(const v2f*)(pG0 + k0);
    v2f wg1 = *(const v2f*)(pG1 + k0);
    u00 = WMMA4(a0, wu0, u00);
    u01 = WMMA4(a0, wu1, u01);
    u10 = WMMA4(a1, wu0, u10);
    u11 = WMMA4(a1, wu1, u11);
    g00 = WMMA4(a0, wg0, g00);
    g01 = WMMA4(a0, wg1, g01);
    g10 = WMMA4(a1, wg0, g10);
    g11 = WMMA4(a1, wg1, g11);
  }

  const float bu0v = bu[n0 + l],      bg0v = bg[n0 + l];
  const float bu1v = bu[n0 + 16 + l], bg1v = bg[n0 + 16 + l];
  const int col0 = n0 + l, col1 = n0 + 16 + l;

#pragma unroll
  for (int r = 0; r < 8; ++r) {
    const int row0 = m0 + 8 * hi + r;   // accumulator half-rows (M = r + 8*hi)
    const int row1 = row0 + 16;
    U[(size_t)row0 * kH + col0] = gate_act(u00[r] + bu0v, g00[r] + bg0v);
    U[(size_t)row0 * kH + col1] = gate_act(u01[r] + bu1v, g01[r] + bg1v);
    U[(size_t)row1 * kH + col0] = gate_act(u10[r] + bu0v, g10[r] + bg0v);
    U[(size_t)row1 * kH + col1] = gate_act(u11[r] + bu1v, g11[r] + bg1v);
  }
}

// ---------------------------------------------------------------------------
// Kernels 2 & 5: generic  Out[m,n] = sum_k Aop[m,k] * W[n,k]  (+ bias[n])
// ---------------------------------------------------------------------------
template <bool BIAS>
__global__ __launch_bounds__(256) void k_gemm_nt(
    const float* __restrict__ Aop, const float* __restrict__ W,
    const float* __restrict__ bias, float* __restrict__ Out) {
  const int wave = (blockIdx.x * blockDim.x + threadIdx.x) >> 5;
  const int lane = threadIdx.x & 31;
  const int tilesM = kM / 32;
  const int tn = wave / tilesM;
  const int tm = wave % tilesM;
  const int m0 = tm * 32, n0 = tn * 32;
  const int l = lane & 15, hi = lane >> 4;

  const float* pA0 = Aop + (size_t)(m0 + l) * kD + 2 * hi;
  const float* pA1 = pA0 + (size_t)16 * kD;
  const float* pB0 = W + (size_t)(n0 + l) * kD + 2 * hi;
  const float* pB1 = pB0 + (size_t)16 * kD;

  v8f c00 = {}, c01 = {}, c10 = {}, c11 = {};

#pragma unroll 8
  for (int k0 = 0; k0 < kD; k0 += 4) {
    v2f a0 = *(const v2f*)(pA0 + k0);
    v2f a1 = *(const v2f*)(pA1 + k0);
    v2f b0 = *(const v2f*)(pB0 + k0);
    v2f b1 = *(const v2f*)(pB1 + k0);
    c00 = WMMA4(a0, b0, c00);
    c01 = WMMA4(a0, b1, c01);
    c10 = WMMA4(a1, b0, c10);
    c11 = WMMA4(a1, b1, c11);
  }

  const int col0 = n0 + l, col1 = n0 + 16 + l;
  float bias0 = 0.f, bias1 = 0.f;
  if (BIAS) { bias0 = bias[col0]; bias1 = bias[col1]; }

#pragma unroll
  for (int r = 0; r < 8; ++r) {
    const int row0 = m0 + 8 * hi + r;
    const int row1 = row0 + 16;
    Out[(size_t)row0 * kH + col0] = c00[r] + bias0;
    Out[(size_t)row0 * kH + col1] = c01[r] + bias1;
    Out[(size_t)row1 * kH + col0] = c10[r] + bias0;
    Out[(size_t)row1 * kH + col1] = c11[r] + bias1;
  }
}

// ---------------------------------------------------------------------------
// Kernel 3: chunked scan. One thread per (b, chunk c, channel h).
//   Local scan with s_{-1}=0 over L=128 steps; chunk-end value -> E[(b*C+c)*H+h].
//   A is diagonal (Mamba/S4D); decay a_h read from A's diagonal.
// ---------------------------------------------------------------------------
__global__ __launch_bounds__(256) void k_scan_chunk(
    const float* __restrict__ uB, const float* __restrict__ Am,
    float* __restrict__ S, float* __restrict__ E) {
  const int idx = blockIdx.x * blockDim.x + threadIdx.x;  // [0, B*C*H)
  const int h = idx % kH;
  const int c = (idx / kH) % kC;
  const int b = idx / (kH * kC);
  const float a = Am[(size_t)h * kH + h];  // diagonal decay

  const size_t base = ((size_t)b * kT + (size_t)c * kL) * kH + h;
  float s = 0.f;
#pragma unroll 4
  for (int k = 0; k < kL; ++k) {
    s = fmaf(a, s, uB[base + (size_t)k * kH]);
    S[base + (size_t)k * kH] = s;
  }
  E[idx] = s;
}

// ---------------------------------------------------------------------------
// Kernel 4: carry fixup.  carry_in(c) = sum_{j<c} a^{L*(c-1-j)} * E_j ;
//   S[t_local=k] += a^{k+1} * carry_in(c).
// ---------------------------------------------------------------------------
__global__ __launch_bounds__(256) void k_scan_fix(
    const float* __restrict__ Am, float* __restrict__ S,
    const float* __restrict__ E) {
  const int idx = blockIdx.x * blockDim.x + threadIdx.x;
  const int h = idx % kH;
  const int c = (idx / kH) % kC;
  const int b = idx / (kH * kC);
  if (c == 0) return;

  const float a = Am[(size_t)h * kH + h];
  float aL = a;                       // a^128 via 7 squarings
#pragma unroll
  for (int q = 0; q < 7; ++q) aL *= aL;

  float carry = 0.f;
  for (int j = 0; j < c; ++j) carry = fmaf(aL, carry, E[((size_t)b * kC + j) * kH + h]);

  const size_t base = ((size_t)b * kT + (size_t)c * kL) * kH + h;
  float f = a;
#pragma unroll 4
  for (int k = 0; k < kL; ++k) {
    S[base + (size_t)k * kH] = fmaf(f, carry, S[base + (size_t)k * kH]);
    f *= a;
  }
}

// ---------------------------------------------------------------------------
extern "C" void kernel_launch(void* const* d_in, const int* in_sizes, int n_in,
                              void* d_out, int out_size, void* d_ws, size_t ws_size,
                              hipStream_t stream) {
  const float* x  = (const float*)d_in[0];
  const float* Wu = (const float*)d_in[1];
  const float* bu = (const float*)d_in[2];
  const float* Wg = (const float*)d_in[3];
  const float* bg = (const float*)d_in[4];
  const float* Am = (const float*)d_in[5];
  const float* Bm = (const float*)d_in[6];
  const float* Cm = (const float*)d_in[7];
  const float* Dv = (const float*)d_in[8];
  float* out = (float*)d_out;

  char* ws = (char*)d_ws;
  const size_t bytesMH = (size_t)kM * kH * sizeof(float);  // 32 MB
  float* U  = (float*)(ws);                 // region 0: u (dead after K2)
  float* uB = (float*)(ws + bytesMH);       // region 1: uB
  float* S  = U;                            // region 0 reused for S
  float* E  = (float*)(ws + 2 * bytesMH);   // 256 KB of chunk carries

  const int gemmBlocks = (kM / 32) * (kH / 32) / 8;   // 8 waves / block -> 1024
  const int scanBlocks = (kB * kC * kH) / 256;        // 65536 threads -> 256

  k_gate_gemm<<<gemmBlocks, 256, 0, stream>>>(x, Wu, bu, Wg, bg, U);
  k_gemm_nt<false><<<gemmBlocks, 256, 0, stream>>>(U, Bm, nullptr, uB);
  k_scan_chunk<<<scanBlocks, 256, 0, stream>>>(uB, Am, S, E);
  k_scan_fix<<<scanBlocks, 256, 0, stream>>>(Am, S, E);
  k_gemm_nt<true><<<gemmBlocks, 256, 0, stream>>>(S, Cm, Dv, out);
}